// Encoder_44994077393185
// MI455X (gfx1250) — compile-verified
//
#include <hip/hip_runtime.h>
#include <math.h>

typedef float v2f __attribute__((ext_vector_type(2)));
typedef float v8f __attribute__((ext_vector_type(8)));

// ---------------------------------------------------------------------------
// CDNA5 async global->LDS copy (16 bytes per lane), tracked by ASYNCcnt.
// VDST = VGPR holding the LDS byte offset (low 32 bits of the generic
// pointer are the LDS byte address per ISA 10.2 aperture rules),
// VADDR = 64-bit global address.
// ---------------------------------------------------------------------------
__device__ __forceinline__ void async_copy_b128(const float* g, float* l) {
    const unsigned loff = (unsigned)(unsigned long long)l;
    asm volatile("global_load_async_to_lds_b128 %0, %1, off"
                 :: "v"(loff), "v"(g)
                 : "memory");
}
#define WAIT_ASYNC(n) asm volatile("s_wait_asynccnt %0" :: "i"(n) : "memory")

// ---------------------------------------------------------------------------
// Token recovery from one-hot rows: tok[t] = index where X[t, v] == 1
// ---------------------------------------------------------------------------
__global__ __launch_bounds__(256) void find_token(const float* __restrict__ X,
                                                  int* __restrict__ tok, int V) {
    __shared__ int s;
    const int t = blockIdx.x;
    if (threadIdx.x == 0) s = 0;
    __syncthreads();
    const float* row = X + (long long)t * V;
    for (int v = threadIdx.x; v < V; v += blockDim.x)
        if (row[v] > 0.5f) atomicMax(&s, v);
    __syncthreads();
    if (threadIdx.x == 0) tok[t] = s;
}

// Xe[t, :] = W_emb[tok[t], :] + pos[t, :]
__global__ __launch_bounds__(256) void embed_add(const int* __restrict__ tok,
                                                 const float* __restrict__ W_emb,
                                                 const float* __restrict__ pos,
                                                 float* __restrict__ Xe, int D) {
    const int t = blockIdx.x;
    const float* wr = W_emb + (long long)tok[t] * D;
    const float* pr = pos + (long long)t * D;
    float* o = Xe + (long long)t * D;
    for (int d = threadIdx.x; d < D; d += blockDim.x) o[d] = wr[d] + pr[d];
}

// ---------------------------------------------------------------------------
// LayerNorm over the SEQUENCE axis (per feature column d), unbiased std.
// Block = 64 columns x 4 row-strips; LDS reduction of the 4 partials.
// ---------------------------------------------------------------------------
__global__ __launch_bounds__(256) void layernorm_seq(const float* __restrict__ in,
                                                     float* __restrict__ out,
                                                     const float* __restrict__ g,
                                                     const float* __restrict__ b,
                                                     int T, int D) {
    __shared__ float ssum[4][64], ssq[4][64];
    __shared__ float smu[64], sginv[64], sbb[64];
    const int tx = threadIdx.x & 63;
    const int ty = threadIdx.x >> 6;
    const int d = blockIdx.x * 64 + tx;
    float sum = 0.f, sq = 0.f;
    for (int t = ty; t < T; t += 4) {
        const float x = in[(long long)t * D + d];
        sum += x; sq += x * x;
    }
    ssum[ty][tx] = sum; ssq[ty][tx] = sq;
    __syncthreads();
    if (ty == 0) {
        sum = ssum[0][tx] + ssum[1][tx] + ssum[2][tx] + ssum[3][tx];
        sq  = ssq[0][tx] + ssq[1][tx] + ssq[2][tx] + ssq[3][tx];
        const float mu  = sum / (float)T;
        const float var = (sq - sum * mu) / (float)(T - 1);
        const float inv = rsqrtf(var);
        smu[tx] = mu; sginv[tx] = g[d] * inv; sbb[tx] = b[d];
    }
    __syncthreads();
    const float mu = smu[tx], gg = sginv[tx], bb = sbb[tx];
    for (int t = ty; t < T; t += 4) {
        const float x = in[(long long)t * D + d];
        out[(long long)t * D + d] = bb + gg * (x - mu);
    }
}

// ---------------------------------------------------------------------------
// Row softmax: one block per row of length n
// ---------------------------------------------------------------------------
__global__ __launch_bounds__(256) void softmax_rows(float* __restrict__ S, int n) {
    __shared__ float red[256];
    float* p = S + (long long)blockIdx.x * n;
    const int tid = threadIdx.x;
    float m = -3.4e38f;
    for (int c = tid; c < n; c += 256) m = fmaxf(m, p[c]);
    red[tid] = m; __syncthreads();
    for (int s = 128; s > 0; s >>= 1) {
        if (tid < s) red[tid] = fmaxf(red[tid], red[tid + s]);
        __syncthreads();
    }
    m = red[0]; __syncthreads();
    float sum = 0.f;
    for (int c = tid; c < n; c += 256) {
        const float e = __expf(p[c] - m);
        p[c] = e; sum += e;
    }
    red[tid] = sum; __syncthreads();
    for (int s = 128; s > 0; s >>= 1) {
        if (tid < s) red[tid] += red[tid + s];
        __syncthreads();
    }
    const float inv = 1.0f / red[0];
    for (int c = tid; c < n; c += 256) p[c] *= inv;
}

// ---------------------------------------------------------------------------
// fp32 WMMA GEMM body: C = alpha * A(MxK) * B(KxN) [+bias] [+Res] [relu]
//   Block tile = (64*MT) x 64; 8 wave32; wave = MT m-tiles x 2 n-tiles.
//   BT=1: B stored (N x K) -> async direct copy to N-major LDS
//   BT=0: B stored (K x N) -> sync loads, reg-double-buffered, transposed
//          into N-major LDS after the compute phase (latency hidden).
//   A tiles staged with GLOBAL_LOAD_ASYNC_TO_LDS_B128 + ping-pong LDS;
//   K loop manually 2x-unrolled (K multiple of 32) with literal buffer
//   indices + peeled final tile -> branch-free steady state.
// LDS rows padded to 20 floats: 16B-aligned b128 async destinations and
// conflict-free b64 fragment reads.
// Fragment layouts per CDNA5 ISA 7.12.2 (A/B lane-symmetric, m<->n):
//   A(16x4): lane = m + 16*(k/2), vgpr = k&1 -> contiguous b64 at la[m][k]
//   B(4x16): lane = n + 16*(k/2), vgpr = k&1 -> contiguous b64 at lbT[n][k]
//   D(16x16): m = vgpr + 8*(lane>=16), n = lane&15
// ---------------------------------------------------------------------------
template <int MT, bool BT, bool BIAS, bool RELU, bool RES>
__device__ __forceinline__ void gemm_body(
    const float* __restrict__ Ab, int lda,
    const float* __restrict__ Bb, int ldb,
    float* __restrict__ Cb, int ldc,
    const float* __restrict__ bias,
    const float* __restrict__ Res, int ldr,
    float alpha, int K) {

    __shared__ __align__(16) float la[2][64 * MT * 20];
    __shared__ __align__(16) float lb[2][64 * 20];   // N-major: lb[..][n*20 + k]

    constexpr int NA = MT + (BT ? 1 : 0);  // async ops per wave per tile

    const int tid = threadIdx.x;
    const int row0 = blockIdx.y * (64 * MT);
    const int col0 = blockIdx.x * 64;

    const int wave = tid >> 5;
    const int lane = tid & 31;
    const int mg  = wave & 3;          // wave's m-group (MT tiles each)
    const int tn0 = (wave >> 2) * 2;   // wave's first N tile (0 or 2)

    v8f acc[MT][2];
    const v8f zero = {0.f, 0.f, 0.f, 0.f, 0.f, 0.f, 0.f, 0.f};
#pragma unroll
    for (int i = 0; i < MT; ++i) { acc[i][0] = zero; acc[i][1] = zero; }

    const int ar  = tid >> 2, ac  = (tid & 3) * 4;   // A / B(BT) staging coords
    const int bkr = tid >> 4, bcc = (tid & 15) * 4;  // B(!BT) staging coords

    const int ml  = lane & 15;
    const int kq2 = (lane >> 4) * 2;
    const int nc  = tn0 * 16 + ml;

    const int ntiles = K >> 4;   // even by construction (K multiple of 32)
    float4 rb{};                 // register-staged B slice (!BT only)

    auto issue = [&](int kt, int bf) {
        const int k0 = kt * 16;
#pragma unroll
        for (int s = 0; s < MT; ++s)
            async_copy_b128(Ab + (long long)(row0 + ar + 64 * s) * lda + (k0 + ac),
                            &la[bf][(ar + 64 * s) * 20 + ac]);
        if (BT)
            async_copy_b128(Bb + (long long)(col0 + ar) * ldb + (k0 + ac),
                            &lb[bf][ar * 20 + ac]);
        else
            rb = *(const float4*)(Bb + (long long)(k0 + bkr) * ldb + (col0 + bcc));
    };
    auto commit = [&](int bf) {
        lb[bf][(bcc + 0) * 20 + bkr] = rb.x;
        lb[bf][(bcc + 1) * 20 + bkr] = rb.y;
        lb[bf][(bcc + 2) * 20 + bkr] = rb.z;
        lb[bf][(bcc + 3) * 20 + bkr] = rb.w;
    };
    auto compute = [&](int cur) {
        const float* lac = &la[cur][0];
        const float* lbc = &lb[cur][0];
#pragma unroll
        for (int kk = 0; kk < 4; ++kk) {
            const int k = kk * 4 + kq2;
            const v2f b0 = *(const v2f*)&lbc[nc * 20 + k];
            const v2f b1 = *(const v2f*)&lbc[(nc + 16) * 20 + k];
#pragma unroll
            for (int i = 0; i < MT; ++i) {
                const v2f a = *(const v2f*)&lac[(mg * 16 * MT + i * 16 + ml) * 20 + k];
                acc[i][0] = __builtin_amdgcn_wmma_f32_16x16x4_f32(
                    false, a, false, b0, (short)0, acc[i][0], false, false);
                acc[i][1] = __builtin_amdgcn_wmma_f32_16x16x4_f32(
                    false, a, false, b1, (short)0, acc[i][1], false, false);
            }
        }
    };
    auto step = [&](int kt, int cur) {
        const int nxt = cur ^ 1;
        issue(kt + 1, nxt);              // prefetch next tile (async / regs)
        WAIT_ASYNC(NA);                  // current tile's asyncs complete
        __syncthreads();
        compute(cur);
        if (!BT) commit(nxt);            // store reg-staged B after compute
        __syncthreads();
    };

    // prologue: stage tile 0 into buffer 0
    issue(0, 0);
    if (!BT) commit(0);

    // steady state, 2x unrolled so LDS buffer offsets are literals
    for (int kt = 0; kt < ntiles - 2; kt += 2) { step(kt, 0); step(kt + 1, 1); }
    step(ntiles - 2, 0);
    // final tile (buffer 1), nothing left to prefetch
    WAIT_ASYNC(0);
    __syncthreads();
    compute(1);

    const int mh = (lane >> 4) * 8;
#pragma unroll
    for (int i = 0; i < MT; ++i) {
#pragma unroll
        for (int j = 0; j < 2; ++j) {
            const int col = col0 + (tn0 + j) * 16 + ml;
#pragma unroll
            for (int r = 0; r < 8; ++r) {
                const int row = row0 + mg * 16 * MT + i * 16 + mh + r;
                float v = acc[i][j][r] * alpha;
                if (BIAS) v += bias[col];
                if (RES)  v += Res[(long long)row * ldr + col];
                if (RELU) v = fmaxf(v, 0.f);
                Cb[(long long)row * ldc + col] = v;
            }
        }
    }
}

// Generic batched wrapper
template <int MT, bool BT, bool BIAS, bool RELU, bool RES>
__global__ __launch_bounds__(256) void wmma_gemm(
    const float* __restrict__ A, long long sA, int lda,
    const float* __restrict__ B, long long sB, int ldb,
    float* __restrict__ C, long long sC, int ldc,
    const float* __restrict__ bias,
    const float* __restrict__ Res, int ldr,
    float alpha, int K) {
    const long long bz = blockIdx.z;
    gemm_body<MT, BT, BIAS, RELU, RES>(A + bz * sA, lda, B + bz * sB, ldb,
                                       C + bz * sC, ldc, bias, Res, ldr, alpha, K);
}

// Fused Q/K/V projection wrapper: grid.z = 3*H; z>>4 selects the matrix,
// z&15 the head (H == 16).
template <int MT>
__global__ __launch_bounds__(256) void wmma_gemm_qkv(
    const float* __restrict__ Yn,
    const float* __restrict__ WQ, const float* __restrict__ WK,
    const float* __restrict__ WV,
    float* __restrict__ Q, float* __restrict__ Kb, float* __restrict__ Vb,
    int D, int HD, int T) {
    const int z = blockIdx.z;
    const int which = z >> 4;
    const int h = z & 15;
    const float* B = (which == 0) ? WQ : (which == 1) ? WK : WV;
    float* C = (which == 0) ? Q : (which == 1) ? Kb : Vb;
    gemm_body<MT, false, false, false, false>(
        Yn, D,
        B + (long long)h * D * HD, HD,
        C + (long long)h * T * HD, HD,
        nullptr, nullptr, 0, 1.f, D);
}

// ---------------------------------------------------------------------------
// Host orchestration
// ---------------------------------------------------------------------------
extern "C" void kernel_launch(void* const* d_in, const int* in_sizes, int n_in,
                              void* d_out, int out_size, void* d_ws, size_t ws_size,
                              hipStream_t stream) {
    (void)in_sizes; (void)n_in; (void)out_size; (void)ws_size;
    constexpr int Dn = 1024, Hn = 16, HDn = 64, FF = 4096, Vn = 32000, Tn = 1024, Ln = 6;

    const float* X      = (const float*)d_in[0];
    const float* W_emb  = (const float*)d_in[1];
    const float* pos    = (const float*)d_in[2];
    const float* WQ     = (const float*)d_in[3];
    const float* WK     = (const float*)d_in[4];
    const float* WV     = (const float*)d_in[5];
    const float* Wout   = (const float*)d_in[6];
    const float* ln1_g  = (const float*)d_in[7];
    const float* ln1_b  = (const float*)d_in[8];
    const float* ln2_g  = (const float*)d_in[9];
    const float* ln2_b  = (const float*)d_in[10];
    const float* w1     = (const float*)d_in[11];
    const float* b1     = (const float*)d_in[12];
    const float* w2     = (const float*)d_in[13];
    const float* b2     = (const float*)d_in[14];
    const float* last_g = (const float*)d_in[15];
    const float* last_b = (const float*)d_in[16];
    float* out = (float*)d_out;

    // Workspace layout (floats), ~88.1 MB total:
    //   6 * T*D activation buffers + H*T*T score buffer (+ 1024 ints).
    float* ws = (float*)d_ws;
    const size_t TD = (size_t)Tn * Dn;
    float* Xe = ws;
    float* Y  = Xe + TD;
    float* Yn = Y + TD;
    float* Q  = Yn + TD;
    float* Kb = Q + TD;
    float* Vb = Kb + TD;
    float* S  = Vb + TD;            // H*T*T floats
    float* Xo = Q;                  // alias: Q dead after the score GEMM
    float* Hb = S;                  // alias: S dead after A@V
    int* tok = (int*)(S + (size_t)Hn * Tn * Tn);

    find_token<<<Tn, 256, 0, stream>>>(X, tok, Vn);
    embed_add<<<Tn, 256, 0, stream>>>(tok, W_emb, pos, Xe, Dn);

    const float inv_sqrt_hd = 0.125f;  // 1/sqrt(64)
    constexpr int MT = 2;              // block tile = 128 x 64

    for (int l = 0; l < Ln; ++l) {
        const float* WQl = WQ + (size_t)l * Hn * Dn * HDn;
        const float* WKl = WK + (size_t)l * Hn * Dn * HDn;
        const float* WVl = WV + (size_t)l * Hn * Dn * HDn;
        const float* Wol = Wout + (size_t)l * Dn * Dn;
        const float* w1l = w1 + (size_t)l * FF * Dn;
        const float* w2l = w2 + (size_t)l * Dn * FF;

        // LN1 (over sequence axis)
        layernorm_seq<<<Dn / 64, 256, 0, stream>>>(Xe, Yn, ln1_g + l * Dn, ln1_b + l * Dn, Tn, Dn);

        // Fused Q/K/V projections: one launch, grid.z = 3*H
        wmma_gemm_qkv<MT><<<dim3(1, Tn / (64 * MT), 3 * Hn), 256, 0, stream>>>(
            Yn, WQl, WKl, WVl, Q, Kb, Vb, Dn, HDn, Tn);

        // Scores: Z = Q @ K^T / sqrt(HD) (B N-major -> async path), per head
        wmma_gemm<MT, true, false, false, false>
            <<<dim3(Tn / 64, Tn / (64 * MT), Hn), 256, 0, stream>>>(
            Q, (long long)Tn * HDn, HDn, Kb, (long long)Tn * HDn, HDn,
            S, (long long)Tn * Tn, Tn, nullptr, nullptr, 0, inv_sqrt_hd, HDn);

        softmax_rows<<<Hn * Tn, 256, 0, stream>>>(S, Tn);

        // R = A @ V, written directly into concatenated (T, D) layout:
        // per-head column offset via batch stride HD, row stride D.
        wmma_gemm<MT, false, false, false, false>
            <<<dim3(HDn / 64, Tn / (64 * MT), Hn), 256, 0, stream>>>(
            S, (long long)Tn * Tn, Tn, Vb, (long long)Tn * HDn, HDn,
            Xo, (long long)HDn, Dn, nullptr, nullptr, 0, 1.f, Tn);

        // Out projection + residual: Y = Xe + Xo @ Wout
        wmma_gemm<MT, false, false, false, true>
            <<<dim3(Dn / 64, Tn / (64 * MT), 1), 256, 0, stream>>>(
            Xo, 0, Dn, Wol, 0, Dn, Y, 0, Dn, nullptr, Xe, Dn, 1.f, Dn);

        // LN2
        layernorm_seq<<<Dn / 64, 256, 0, stream>>>(Y, Yn, ln2_g + l * Dn, ln2_b + l * Dn, Tn, Dn);

        // FFN1: Hb = relu(Yn @ w1^T + b1)   (w1 is (DFF, D) -> async BT path)
        wmma_gemm<MT, true, true, true, false>
            <<<dim3(FF / 64, Tn / (64 * MT), 1), 256, 0, stream>>>(
            Yn, 0, Dn, w1l, 0, Dn, Hb, 0, FF, b1 + (size_t)l * FF, nullptr, 0, 1.f, Dn);

        // FFN2: Xe = Y + Hb @ w2^T + b2     (w2 is (D, DFF) -> async BT path)
        wmma_gemm<MT, true, true, false, true>
            <<<dim3(Dn / 64, Tn / (64 * MT), 1), 256, 0, stream>>>(
            Hb, 0, FF, w2l, 0, FF, Xe, 0, Dn, b2 + (size_t)l * Dn, Y, Dn, 1.f, FF);
    }

    // Final LayerNorm straight into d_out
    layernorm_seq<<<Dn / 64, 256, 0, stream>>>(Xe, out, last_g, last_b, Tn, Dn);
}